// GNN_14465449853446
// MI455X (gfx1250) — compile-verified
//
#include <hip/hip_runtime.h>

#define HID 128
#define LDSS 132   // padded LDS row stride (floats) -> conflict-free frag reads

typedef __attribute__((ext_vector_type(2))) float v2f;
typedef __attribute__((ext_vector_type(8))) float v8f;

// ---------------- zero ----------------
__global__ void gnn_zero_f32(float* __restrict__ p, int n) {
    int t = blockIdx.x * blockDim.x + threadIdx.x;
    if (t < n) p[t] = 0.0f;
}

// ---------------- degree count ----------------
__global__ void gnn_count_deg(const int* __restrict__ dst, float* __restrict__ deg, int nE) {
    int e = blockIdx.x * blockDim.x + threadIdx.x;
    if (e < nE) atomicAdd(&deg[dst[e]], 1.0f);
}

// ---------------- edge scatter-add: agg[dst] += feat[src] ----------------
// thread t -> edge t>>7, feature t&127 : lanes of a wave32 cover 32 consecutive
// features of one edge -> fully coalesced 128B gathers + atomics (L2-resident).
__global__ void gnn_scatter(const float* __restrict__ feat, const int* __restrict__ src,
                            const int* __restrict__ dst, float* __restrict__ agg, int nE) {
    int t = blockIdx.x * 256 + threadIdx.x;
    int e = t >> 7;
    if (e >= nE) return;
    int f = t & 127;
    atomicAdd(&agg[dst[e] * HID + f], feat[src[e] * HID + f]);
}

// ---------------- mean: agg /= max(deg,1) ----------------
__global__ void gnn_mean(float* __restrict__ agg, const float* __restrict__ deg, int nfeat) {
    int t = blockIdx.x * blockDim.x + threadIdx.x;
    if (t >= nfeat) return;
    float d = deg[t >> 7];
    agg[t] *= 1.0f / fmaxf(d, 1.0f);
}

// ---------------- fused SAGE GEMM: out = [relu](agg@Wl^T + bl + xroot@Wr^T) ----
// Block = 8 wave32s, one 16-row node tile; wave w owns output cols [16w,16w+16).
// K-loop of v_wmma_f32_16x16x4_f32, two independent accumulators (agg-path,
// root-path) for ILP; bias+ReLU fused in the epilogue.
template <bool RELU>
__global__ __launch_bounds__(256)
void gnn_sage_gemm(const float* __restrict__ agg, const float* __restrict__ xroot,
                   const float* __restrict__ Wl, const float* __restrict__ bl,
                   const float* __restrict__ Wr, float* __restrict__ out) {
    __shared__ float sA[16 * LDSS];   // agg tile (16 x 128, padded)
    __shared__ float sR[16 * LDSS];   // root tile

    const int n0  = blockIdx.x * 16;
    const int tid = threadIdx.x;

    // Stage both 16x128 A-tiles to LDS (8 floats per thread per tile, coalesced).
    for (int i = tid; i < 16 * HID; i += 256) {
        int r = i >> 7, c = i & 127;
        sA[r * LDSS + c] = agg[(n0 + r) * HID + c];
        sR[r * LDSS + c] = xroot[(n0 + r) * HID + c];
    }
    __syncthreads();

    const int wave  = tid >> 5;
    const int lane  = tid & 31;
    const int o0    = wave * 16;
    const int mrow  = lane & 15;           // A row / B output-col within tile
    const int khalf = (lane >> 4) << 1;    // ISA 16x4 f32 layout: hi half-wave holds K+2,K+3

    const float* wl = Wl + (o0 + mrow) * HID + khalf;  // B[k][o] = W[o][k] -> contiguous k pair
    const float* wr = Wr + (o0 + mrow) * HID + khalf;
    const float* pa = sA + mrow * LDSS + khalf;
    const float* pr = sR + mrow * LDSS + khalf;

    v8f cL = {0.f, 0.f, 0.f, 0.f, 0.f, 0.f, 0.f, 0.f};
    v8f cR = {0.f, 0.f, 0.f, 0.f, 0.f, 0.f, 0.f, 0.f};

#pragma unroll 8
    for (int k = 0; k < HID; k += 4) {
        v2f aA = *(const v2f*)(pa + k);
        v2f aR = *(const v2f*)(pr + k);
        v2f bL = *(const v2f*)(wl + k);
        v2f bR = *(const v2f*)(wr + k);
        cL = __builtin_amdgcn_wmma_f32_16x16x4_f32(false, aA, false, bL,
                                                   (short)0, cL, false, false);
        cR = __builtin_amdgcn_wmma_f32_16x16x4_f32(false, aR, false, bR,
                                                   (short)0, cR, false, false);
    }

    // C/D layout: VGPR r -> row (r + 8*(lane>=16)), col = lane&15.
    const float bias = bl[o0 + mrow];
    float* op = out + (n0 + (lane >> 4) * 8) * HID + o0 + mrow;
#pragma unroll
    for (int r = 0; r < 8; ++r) {
        float v = cL[r] + cR[r] + bias;
        if (RELU) v = fmaxf(v, 0.0f);
        op[r * HID] = v;
    }
}

extern "C" void kernel_launch(void* const* d_in, const int* in_sizes, int n_in,
                              void* d_out, int out_size, void* d_ws, size_t ws_size,
                              hipStream_t stream) {
    const float* x   = (const float*)d_in[0];
    const int*   ei  = (const int*)d_in[1];
    const float* W1l = (const float*)d_in[2];
    const float* b1l = (const float*)d_in[3];
    const float* W1r = (const float*)d_in[4];
    const float* W2l = (const float*)d_in[5];
    const float* b2l = (const float*)d_in[6];
    const float* W2r = (const float*)d_in[7];
    float* out = (float*)d_out;

    const int nN = in_sizes[0] / HID;   // 50000
    const int nE = in_sizes[1] / 2;     // 600000
    const int* src = ei;                // edge_index row 0
    const int* dst = ei + nE;           // edge_index row 1

    const int nfeat = nN * HID;
    // workspace layout: agg[nfeat] | deg[nN] | h[nfeat]  (~51.4 MB)
    float* agg = (float*)d_ws;
    float* deg = agg + (size_t)nfeat;
    float* h   = deg + (size_t)nN;

    const int edgeThreads = nE * HID;   // 76.8M, fits int

    // ---- layer 1 ----
    gnn_zero_f32<<<(nfeat + nN + 255) / 256, 256, 0, stream>>>(agg, nfeat + nN); // agg+deg
    gnn_count_deg<<<(nE + 255) / 256, 256, 0, stream>>>(dst, deg, nE);
    gnn_scatter<<<(edgeThreads + 255) / 256, 256, 0, stream>>>(x, src, dst, agg, nE);
    gnn_mean<<<(nfeat + 255) / 256, 256, 0, stream>>>(agg, deg, nfeat);
    gnn_sage_gemm<true><<<nN / 16, 256, 0, stream>>>(agg, x, W1l, b1l, W1r, h);

    // ---- layer 2 (same graph -> deg reused) ----
    gnn_zero_f32<<<(nfeat + 255) / 256, 256, 0, stream>>>(agg, nfeat);
    gnn_scatter<<<(edgeThreads + 255) / 256, 256, 0, stream>>>(h, src, dst, agg, nE);
    gnn_mean<<<(nfeat + 255) / 256, 256, 0, stream>>>(agg, deg, nfeat);
    gnn_sage_gemm<false><<<nN / 16, 256, 0, stream>>>(agg, h, W2l, b2l, W2r, out);
}